// DifferentiableWalker_41953240547536
// MI455X (gfx1250) — compile-verified
//
#include <hip/hip_runtime.h>
#include <stdint.h>

#define BLOCK 64               // threads per block = 32 walker-pairs
#define PAIRS (BLOCK / 2)
#define TILE  32               // steps staged in LDS per flush
#define ROWF  (TILE * 3 + 4)   // 100 floats/row: 400B stride, 16B aligned

// Adjacent-lane swap via DPP quad_perm [1,0,3,2] (ctrl=0xB1): pure VALU, no LDS.
__device__ __forceinline__ uint32_t swap_pair_u32(uint32_t v) {
  return (uint32_t)__builtin_amdgcn_mov_dpp((int)v, 0xB1, 0xF, 0xF, true);
}
__device__ __forceinline__ float swap_pair_f32(float v) {
  return __int_as_float(
      __builtin_amdgcn_mov_dpp(__float_as_int(v), 0xB1, 0xF, 0xF, true));
}

// ---- JAX threefry2x32 (5 groups of 4 rounds) ----
__device__ __forceinline__ void tf2x32(uint32_t k0, uint32_t k1,
                                       uint32_t& x0, uint32_t& x1) {
  const uint32_t ks2 = k0 ^ k1 ^ 0x1BD11BDAu;
  x0 += k0; x1 += k1;
#define TFR(r) { x0 += x1; x1 = (x1 << (r)) | (x1 >> (32 - (r))); x1 ^= x0; }
  TFR(13) TFR(15) TFR(26) TFR(6)
  x0 += k1;  x1 += ks2 + 1u;
  TFR(17) TFR(29) TFR(16) TFR(24)
  x0 += ks2; x1 += k0 + 2u;
  TFR(13) TFR(15) TFR(26) TFR(6)
  x0 += k0;  x1 += k1 + 3u;
  TFR(17) TFR(29) TFR(16) TFR(24)
  x0 += k1;  x1 += ks2 + 4u;
  TFR(13) TFR(15) TFR(26) TFR(6)
  x0 += ks2; x1 += k0 + 5u;
#undef TFR
}

// ---- erfinv (single precision, Giles 2010) ----
__device__ __forceinline__ float erfinv_f(float x) {
  float w = -__logf((1.0f - x) * (1.0f + x));
  float p;
  if (w < 5.0f) {
    w -= 2.5f;
    p =              2.81022636e-08f;
    p = fmaf(p, w,   3.43273939e-07f);
    p = fmaf(p, w,  -3.5233877e-06f);
    p = fmaf(p, w,  -4.39150654e-06f);
    p = fmaf(p, w,   0.00021858087f);
    p = fmaf(p, w,  -0.00125372503f);
    p = fmaf(p, w,  -0.00417768164f);
    p = fmaf(p, w,   0.246640727f);
    p = fmaf(p, w,   1.50140941f);
  } else {
    w = sqrtf(w) - 3.0f;
    p =             -0.000200214257f;
    p = fmaf(p, w,   0.000100950558f);
    p = fmaf(p, w,   0.00134934322f);
    p = fmaf(p, w,  -0.00367342844f);
    p = fmaf(p, w,   0.00573950773f);
    p = fmaf(p, w,  -0.0076224613f);
    p = fmaf(p, w,   0.00943887047f);
    p = fmaf(p, w,   1.00167406f);
    p = fmaf(p, w,   2.83297682f);
  }
  return p * x;
}

// JAX uniform-in-[lo,1) -> N(0,1):  sqrt(2)*erfinv(u), lo = nextafter(-1,0)
__device__ __forceinline__ float bits_to_normal(uint32_t bits) {
  float f = __uint_as_float((bits >> 9) | 0x3f800000u) - 1.0f;   // [0,1)
  float u = fmaf(f, 1.99999994f, -0.99999994f);
  u = fmaxf(u, -0.99999994f);
  return 1.41421356237f * erfinv_f(u);
}

// One walker step, cooperatively on a lane pair.
// lb = 0 (even lane) / 1 (odd lane). Even lane of the pair owns the output.
__device__ __forceinline__ void do_step_pair(uint32_t lb, bool even,
                                             uint32_t& k0, uint32_t& k1,
                                             float sigma_dt,
                                             float& px, float& py, float& pz) {
  // split(key): counters x0s=[0,1], x1s=[2,3]; even lane -> (0,2), odd -> (1,3)
  uint32_t x0 = lb, x1 = 2u + lb;
  tf2x32(k0, k1, x0, x1);
  const uint32_t p0 = swap_pair_u32(x0);
  const uint32_t p1 = swap_pair_u32(x1);
  // step_key = (a0, b0), next_key = (a1, b1)   (a = even-lane block, b = odd)
  const uint32_t sk0 = even ? x0 : p0;
  const uint32_t sk1 = even ? p0 : x0;
  k0 = even ? x1 : p1;
  k1 = even ? p1 : x1;
  // random_bits(step_key,(3,)): counters padded [0,1,2,0] -> pairs (0,2) & (1,0)
  uint32_t y0b = lb, y1b = 2u - 2u * lb;
  tf2x32(sk0, sk1, y0b, y1b);
  const float nA  = bits_to_normal(y0b);       // even: n0, odd: n1
  const float nB  = bits_to_normal(y1b);       // even: n2, odd: unused
  const float nAo = swap_pair_f32(nA);
  const float nBo = swap_pair_f32(nB);
  const float n0 = even ? nA  : nAo;
  const float n1 = even ? nAo : nA;
  const float n2 = even ? nB  : nBo;
  // both lanes track pos redundantly (cheaper than EXEC-masked region)
  px = fmaf(sigma_dt, n0, px);
  py = fmaf(sigma_dt, n1, py);
  pz = fmaf(sigma_dt, n2, pz);
  // reflect if ||p|| > 1:  2/dist - 1 == 2*rsqrt(d2) - 1,  dist>1 <=> d2>1
  const float d2 = px * px + py * py + pz * pz;
  const float sc = (d2 > 1.0f) ? fmaf(2.0f, __frsqrt_rn(d2), -1.0f) : 1.0f;
  px *= sc; py *= sc; pz *= sc;
}

__global__ __launch_bounds__(BLOCK) void walker_kernel(
    const float* __restrict__ y0,
    const float* __restrict__ diffusivity,
    const float* __restrict__ radius,
    const float* __restrict__ dt,
    const uint32_t* __restrict__ keys,
    float* __restrict__ out,
    int nWalkers, int numSteps) {
  __shared__ float tile[2][PAIRS][ROWF];   // double-buffered staging rows

  const int tid  = threadIdx.x;
  const uint32_t lb = (uint32_t)(tid & 1);
  const bool even = (lb == 0u);
  const int pair = tid >> 1;
  const int w    = (blockIdx.x * BLOCK + tid) >> 1;   // 2 lanes per walker
  if (w >= nWalkers) return;

  const float R        = radius[0];
  const float Dunit    = diffusivity[0] / (R * R);
  const float sigma_dt = sqrtf(2.0f * Dunit) * sqrtf(dt[0]);

  float px = y0[w * 3 + 0] / R;
  float py = y0[w * 3 + 1] / R;
  float pz = y0[w * 3 + 2] / R;
  uint32_t k0 = keys[w * 2 + 0];
  uint32_t k1 = keys[w * 2 + 1];

  float* outW = out + (size_t)w * (size_t)numSteps * 3;
  // 32-bit LDS byte offsets of this pair's two staging rows
  const uint32_t ldsRow0 = (uint32_t)(uintptr_t)(&tile[0][pair][0]);
  const uint32_t ldsRow1 = (uint32_t)(uintptr_t)(&tile[1][pair][0]);

  int s = 0;
  const int nTiles = numSteps / TILE;
  for (int t = 0; t < nTiles; ++t) {
    const int buf = t & 1;
    // Double-buffered: allow 6 of last tile's 24 async issues in flight; in-order
    // completion => anything older (same-parity buffer) has drained.
    asm volatile("s_wait_asynccnt 6" ::: "memory");
    for (int i = 0; i < TILE; ++i) {
      do_step_pair(lb, even, k0, k1, sigma_dt, px, py, pz);
      if (even) {
        tile[buf][pair][i * 3 + 0] = px * R;
        tile[buf][pair][i * 3 + 1] = py * R;
        tile[buf][pair][i * 3 + 2] = pz * R;
      }
    }
    // Commit LDS writes before the async DMA engine reads them.
    asm volatile("s_wait_dscnt 0" ::: "memory");
    // Even lane streams the pair's 96 floats (24 x B128) LDS -> global.
    // IOFFSET applies to both LDS and global sides (CDNA5 ISA).
    if (even) {
      const uint64_t g = (uint64_t)(uintptr_t)(outW + (size_t)s * 3);
      const uint32_t l = buf ? ldsRow1 : ldsRow0;
      #pragma unroll
      for (int jj = 0; jj < (TILE * 3) / 4; ++jj) {
        asm volatile("global_store_async_from_lds_b128 %0, %1, off offset:%c2"
                     :: "v"(g), "v"(l), "i"(jj * 16)
                     : "memory");
      }
    }
    s += TILE;
  }
  // Tail for numSteps not a multiple of TILE (dead for 4096, kept for generality).
  for (; s < numSteps; ++s) {
    do_step_pair(lb, even, k0, k1, sigma_dt, px, py, pz);
    if (even) {
      outW[(size_t)s * 3 + 0] = px * R;
      outW[(size_t)s * 3 + 1] = py * R;
      outW[(size_t)s * 3 + 2] = pz * R;
    }
  }
  asm volatile("s_wait_asynccnt 0" ::: "memory");
}

extern "C" void kernel_launch(void* const* d_in, const int* in_sizes, int n_in,
                              void* d_out, int out_size, void* d_ws, size_t ws_size,
                              hipStream_t stream) {
  (void)n_in; (void)d_ws; (void)ws_size;
  const float*    y0   = (const float*)d_in[0];
  const float*    diff = (const float*)d_in[1];
  const float*    rad  = (const float*)d_in[2];
  const float*    dt   = (const float*)d_in[3];
  const uint32_t* keys = (const uint32_t*)d_in[4];
  // d_in[5] (num_steps) lives in device memory; derive it from out_size instead.
  const int nWalkers = in_sizes[0] / 3;
  const int numSteps = out_size / (nWalkers * 3);

  const int nThreads = 2 * nWalkers;                 // lane pair per walker
  dim3 grid((nThreads + BLOCK - 1) / BLOCK), block(BLOCK);
  walker_kernel<<<grid, block, 0, stream>>>(y0, diff, rad, dt, keys,
                                            (float*)d_out, nWalkers, numSteps);
}